// GCN_15504831938806
// MI455X (gfx1250) — compile-verified
//
#include <hip/hip_runtime.h>

typedef __attribute__((ext_vector_type(2))) float v2f;
typedef __attribute__((ext_vector_type(8))) float v8f;

#define F_IN  512
#define F_H   64
#define F_OUT 20

// ---------------- degree / norm ----------------
__global__ __launch_bounds__(256) void k_init_deg(float* deg, int n) {
  int i = blockIdx.x * 256 + threadIdx.x;
  if (i < n) deg[i] = 1.0f;  // self-loop contributes weight 1
}

__global__ __launch_bounds__(256) void k_deg_edges(const int* __restrict__ dst,
                                                   const float* __restrict__ ew,
                                                   float* __restrict__ deg, int E) {
  int e = blockIdx.x * 256 + threadIdx.x;
  if (e < E) atomicAdd(&deg[dst[e]], ew[e]);
}

__global__ __launch_bounds__(256) void k_dinv(float* deg, int n) {
  int i = blockIdx.x * 256 + threadIdx.x;
  if (i < n) { float d = deg[i]; deg[i] = (d > 0.0f) ? rsqrtf(d) : 0.0f; }
}

// ---------------- layer-1 GEMM: xw = x @ W1, fp32 WMMA 16x16x4 ----------------
// One wave computes a 16-row x 64-col panel. K stepped by 4.
// A 16x4 f32 layout: lanes 0-15 (M=0-15) hold {K0,K1} in v0,v1; lanes 16-31 hold {K2,K3}.
// B 4x16 f32 layout: lanes 0-15 (N=0-15) hold rows {K0,K1}; lanes 16-31 hold {K2,K3}.
__global__ __launch_bounds__(128) void k_gemm1(const float* __restrict__ x,
                                               const float* __restrict__ W1,
                                               float* __restrict__ xw, int n) {
  int wid  = (blockIdx.x * 128 + threadIdx.x) >> 5;
  int lane = threadIdx.x & 31;
  int row0 = wid * 16;
  if (row0 >= n) return;

  int m    = lane & 15;          // row within tile (A) / column within tile (B)
  int koff = (lane >> 4) << 1;   // lanes 0-15 -> K pair {0,1}; lanes 16-31 -> {2,3}
  const float* arow = x + (size_t)(row0 + m) * F_IN + koff;

  v8f c0 = {}, c1 = {}, c2 = {}, c3 = {};
  for (int k = 0; k < F_IN; k += 4) {
    v2f a = *(const v2f*)(arow + k);  // A[m][k+koff], A[m][k+koff+1]
    const float* br0 = W1 + (size_t)(k + koff)     * F_H + m;
    const float* br1 = W1 + (size_t)(k + koff + 1) * F_H + m;
    v2f b0 = { br0[0],  br1[0]  };   // tile cols  0..15
    v2f b1 = { br0[16], br1[16] };   // tile cols 16..31
    v2f b2 = { br0[32], br1[32] };   // tile cols 32..47
    v2f b3 = { br0[48], br1[48] };   // tile cols 48..63
    c0 = __builtin_amdgcn_wmma_f32_16x16x4_f32(false, a, false, b0, (short)0, c0, false, false);
    c1 = __builtin_amdgcn_wmma_f32_16x16x4_f32(false, a, false, b1, (short)0, c1, false, false);
    c2 = __builtin_amdgcn_wmma_f32_16x16x4_f32(false, a, false, b2, (short)0, c2, false, false);
    c3 = __builtin_amdgcn_wmma_f32_16x16x4_f32(false, a, false, b3, (short)0, c3, false, false);
  }

  // C/D layout: VGPR r, lanes 0-15 -> M=r, N=lane; lanes 16-31 -> M=r+8, N=lane-16
  int mbase = row0 + ((lane >> 4) ? 8 : 0);
  int col   = lane & 15;
  #pragma unroll
  for (int r = 0; r < 8; ++r) {
    float* orow = xw + (size_t)(mbase + r) * F_H + col;
    orow[0]  = c0[r];
    orow[16] = c1[r];
    orow[32] = c2[r];
    orow[48] = c3[r];
  }
}

// ---------------- layer-1 aggregation ----------------
// init with bias + self-loop term: h1[i] = b1 + xw[i] * dinv[i]^2
__global__ __launch_bounds__(256) void k_selfinit1(const float* __restrict__ xw,
                                                   const float* __restrict__ dinv,
                                                   const float* __restrict__ b1,
                                                   float* __restrict__ h1, int n) {
  int t = blockIdx.x * 256 + threadIdx.x;
  if (t < n * F_H) {
    int i = t >> 6, f = t & 63;
    float di = dinv[i];
    h1[t] = b1[f] + xw[t] * di * di;
  }
}

// one wave per edge, 64 feats (2 per lane); gathers hit L2-resident xw
__global__ __launch_bounds__(256) void k_edge1(const int* __restrict__ src,
                                               const int* __restrict__ dst,
                                               const float* __restrict__ ew,
                                               const float* __restrict__ dinv,
                                               const float* __restrict__ xw,
                                               float* __restrict__ h1, int E) {
  int e = blockIdx.x * 8 + (threadIdx.x >> 5);
  if (e >= E) return;
  int lane = threadIdx.x & 31;
  int s = src[e], d = dst[e];
  float nrm = dinv[s] * ew[e] * dinv[d];
  const float* xr = xw + (size_t)s * F_H;
  float*       hr = h1 + (size_t)d * F_H;
  atomicAdd(&hr[lane],      xr[lane]      * nrm);
  atomicAdd(&hr[lane + 32], xr[lane + 32] * nrm);
}

// ---------------- layer-2 GEMM (ReLU fused): hw2 = relu(h1) @ W2 ----------------
__global__ __launch_bounds__(256) void k_gemm2(const float* __restrict__ h1,
                                               const float* __restrict__ W2,
                                               float* __restrict__ hw2, int n) {
  __shared__ float w2s[F_H * F_OUT];
  for (int t = threadIdx.x; t < F_H * F_OUT; t += 256) w2s[t] = W2[t];
  __syncthreads();
  int row = blockIdx.x * 256 + threadIdx.x;
  if (row >= n) return;
  float acc[F_OUT];
  #pragma unroll
  for (int c = 0; c < F_OUT; ++c) acc[c] = 0.0f;
  const float* hr = h1 + (size_t)row * F_H;
  for (int k = 0; k < F_H; ++k) {
    float hv = fmaxf(hr[k], 0.0f);
    #pragma unroll
    for (int c = 0; c < F_OUT; ++c) acc[c] += hv * w2s[k * F_OUT + c];
  }
  float* o = hw2 + (size_t)row * F_OUT;
  #pragma unroll
  for (int c = 0; c < F_OUT; ++c) o[c] = acc[c];
}

// ---------------- layer-2 aggregation ----------------
__global__ __launch_bounds__(256) void k_selfinit2(const float* __restrict__ hw2,
                                                   const float* __restrict__ dinv,
                                                   const float* __restrict__ b2,
                                                   float* __restrict__ out, int n) {
  int t = blockIdx.x * 256 + threadIdx.x;
  if (t < n * F_OUT) {
    int i = t / F_OUT, c = t % F_OUT;
    float di = dinv[i];
    out[t] = b2[c] + hw2[t] * di * di;
  }
}

// packed: one thread per (edge, output column) -> 100% lane utilization
__global__ __launch_bounds__(256) void k_edge2(const int* __restrict__ src,
                                               const int* __restrict__ dst,
                                               const float* __restrict__ ew,
                                               const float* __restrict__ dinv,
                                               const float* __restrict__ hw2,
                                               float* __restrict__ out, int E) {
  long long t = (long long)blockIdx.x * 256 + threadIdx.x;
  if (t >= (long long)E * F_OUT) return;
  int e = (int)(t / F_OUT);
  int c = (int)(t - (long long)e * F_OUT);
  int s = src[e], d = dst[e];
  float nrm = dinv[s] * ew[e] * dinv[d];
  atomicAdd(&out[(size_t)d * F_OUT + c], hw2[(size_t)s * F_OUT + c] * nrm);
}

// ---------------- row softmax (in place on d_out) ----------------
__global__ __launch_bounds__(256) void k_softmax(float* __restrict__ out, int n) {
  int i = blockIdx.x * 256 + threadIdx.x;
  if (i >= n) return;
  float* r = out + (size_t)i * F_OUT;
  float mx = r[0];
  #pragma unroll
  for (int c = 1; c < F_OUT; ++c) mx = fmaxf(mx, r[c]);
  float ex[F_OUT];
  float sum = 0.0f;
  #pragma unroll
  for (int c = 0; c < F_OUT; ++c) { ex[c] = expf(r[c] - mx); sum += ex[c]; }
  float inv = 1.0f / sum;
  #pragma unroll
  for (int c = 0; c < F_OUT; ++c) r[c] = ex[c] * inv;
}

extern "C" void kernel_launch(void* const* d_in, const int* in_sizes, int n_in,
                              void* d_out, int out_size, void* d_ws, size_t ws_size,
                              hipStream_t stream) {
  const float* x  = (const float*)d_in[0];
  const int*   ei = (const int*)  d_in[1];
  const float* ew = (const float*)d_in[2];
  const float* W1 = (const float*)d_in[3];
  const float* b1 = (const float*)d_in[4];
  const float* W2 = (const float*)d_in[5];
  const float* b2 = (const float*)d_in[6];
  float* out = (float*)d_out;

  const int n = in_sizes[0] / F_IN;   // 100000
  const int E = in_sizes[1] / 2;      // 3200000
  const int* src = ei;
  const int* dst = ei + E;

  // workspace: deg/dinv [n] | xw [n*64] | h1 [n*64]; hw2 reuses xw region
  float* deg = (float*)d_ws;
  float* xw  = deg + n;
  float* h1  = xw + (size_t)n * F_H;
  float* hw2 = xw;

  const int gb = 256;
  k_init_deg <<<(n + gb - 1) / gb, gb, 0, stream>>>(deg, n);
  k_deg_edges<<<(E + gb - 1) / gb, gb, 0, stream>>>(dst, ew, deg, E);
  k_dinv     <<<(n + gb - 1) / gb, gb, 0, stream>>>(deg, n);

  int waves = (n + 15) / 16;
  int gblks = (waves + 3) / 4;        // 4 waves per 128-thread block
  k_gemm1<<<gblks, 128, 0, stream>>>(x, W1, xw, n);

  k_selfinit1<<<((n * F_H) + gb - 1) / gb, gb, 0, stream>>>(xw, deg, b1, h1, n);
  k_edge1    <<<(E + 7) / 8, gb, 0, stream>>>(src, dst, ew, deg, xw, h1, E);

  k_gemm2    <<<(n + gb - 1) / gb, gb, 0, stream>>>(h1, W2, hw2, n);

  k_selfinit2<<<((n * F_OUT) + gb - 1) / gb, gb, 0, stream>>>(hw2, deg, b2, out, n);
  long long e2threads = (long long)E * F_OUT;
  int e2blocks = (int)((e2threads + gb - 1) / gb);
  k_edge2    <<<e2blocks, gb, 0, stream>>>(src, dst, ew, deg, hw2, out, E);

  k_softmax  <<<(n + gb - 1) / gb, gb, 0, stream>>>(out, n);
}